// WeightedCenterLoss_56521769615883
// MI455X (gfx1250) — compile-verified
//
#include <hip/hip_runtime.h>

typedef float v2f __attribute__((ext_vector_type(2)));
typedef float v8f __attribute__((ext_vector_type(8)));

#define FEAT_DIM     256
#define BATCH        2048
#define NUM_CLASSES  50000
#define TILES        (BATCH / 16)   // 128 waves, one 16-sample tile each

// One wave (32 lanes) per block computes 16 per-sample squared distances via
// V_WMMA_F32_16X16X4_F32 on diff = x - c (diagonal of diff * diff^T), applies
// weight + clamp, and reduces to one partial sum per block.
__global__ __launch_bounds__(32)
void WeightedCenterLoss_tile_kernel(const float* __restrict__ x,
                                    const float* __restrict__ centers,
                                    const float* __restrict__ cw,
                                    const int*   __restrict__ labels,
                                    float*       __restrict__ partial) {
    const int lane  = threadIdx.x;            // 0..31
    const int m0    = blockIdx.x * 16;        // tile base sample
    const int row   = m0 + (lane & 15);       // sample this lane serves
    const int khalf = (lane >> 4) * 2;        // lanes 0-15: K{0,1}; 16-31: K{2,3}

    const int   lbl = labels[row];
    const float* __restrict__ xr = x       + (size_t)row * FEAT_DIM + khalf;
    const float* __restrict__ cr = centers + (size_t)lbl * FEAT_DIM + khalf;

    v8f acc = {0.f, 0.f, 0.f, 0.f, 0.f, 0.f, 0.f, 0.f};

#pragma unroll 8
    for (int k = 0; k < FEAT_DIM; k += 4) {
        v2f a = *(const v2f*)(xr + k);        // x  slice, this lane's (row, K pair)
        v2f b = *(const v2f*)(cr + k);        // center slice (gathered row)
        v2f d = a - b;                        // diff feeds both A and B
        // D = A(16x4) * B(4x16) + C ; diagonal accumulates ||x - c||^2 per row
        acc = __builtin_amdgcn_wmma_f32_16x16x4_f32(
                  /*neg_a=*/false, d, /*neg_b=*/false, d,
                  /*c_mod=*/(short)0, acc, /*reuse_a=*/false, /*reuse_b=*/false);
    }

    // Diagonal extraction: (i,i) at {lane=i, comp=i} for i<8,
    // and {lane=i+16, comp=i-8} (i.e. lanes 24..31, comp=lane-24) for i>=8.
    float v = 0.f;
    const bool haveLo = (lane < 8);
    const bool haveHi = (lane >= 24);
    if (haveLo || haveHi) {
        const int idx = haveLo ? lane : (lane - 24);
        float dsq = acc[0];
#pragma unroll
        for (int j = 1; j < 8; ++j)
            if (idx == j) dsq = acc[j];
        const float w = cw[lbl];              // this lane already owns the right label
        float t = dsq * w;
        t = fminf(fmaxf(t, 1e-12f), 1e12f);   // clamp like the reference
        v = t;
    }

    // wave32 sum of the 16 contributions (other lanes carry 0)
#pragma unroll
    for (int off = 16; off > 0; off >>= 1)
        v += __shfl_down(v, off, 32);

    if (lane == 0)
        partial[blockIdx.x] = v;
}

// Deterministic tree reduction of the 128 per-tile partials + off-mask constant.
__global__ __launch_bounds__(TILES)
void WeightedCenterLoss_final_kernel(const float* __restrict__ partial,
                                     float*       __restrict__ out) {
    __shared__ float red[TILES];
    const int t = threadIdx.x;
    red[t] = partial[t];
    __syncthreads();
#pragma unroll
    for (int s = TILES / 2; s > 0; s >>= 1) {
        if (t < s) red[t] += red[t + s];
        __syncthreads();
    }
    if (t == 0) {
        const float off_mask_const =
            (float)((double)BATCH * (double)(NUM_CLASSES - 1)) * 1e-12f;
        out[0] = (red[0] + off_mask_const) / (float)BATCH;
    }
}

extern "C" void kernel_launch(void* const* d_in, const int* in_sizes, int n_in,
                              void* d_out, int out_size, void* d_ws, size_t ws_size,
                              hipStream_t stream) {
    (void)in_sizes; (void)n_in; (void)out_size; (void)ws_size;

    const float* x       = (const float*)d_in[0];   // (2048, 256) f32
    const float* centers = (const float*)d_in[1];   // (50000, 256) f32
    const float* cw      = (const float*)d_in[2];   // (50000,) f32
    const int*   labels  = (const int*)d_in[3];     // (2048,) int32
    float*       out     = (float*)d_out;           // scalar f32
    float*       partial = (float*)d_ws;            // 128 floats of scratch

    WeightedCenterLoss_tile_kernel<<<TILES, 32, 0, stream>>>(
        x, centers, cw, labels, partial);
    WeightedCenterLoss_final_kernel<<<1, TILES, 0, stream>>>(partial, out);
}